// AttributionCentroidTracker_26207890440396
// MI455X (gfx1250) — compile-verified
//
#include <hip/hip_runtime.h>
#include <hip/hip_bf16.h>

// Problem constants (match reference)
#define BB 512
#define VV 32768
#define CC 16
// MOMENTUM = 1 - 2/1001  ->  (1 - MOMENTUM) = 2/1001
#define ONE_MINUS_M (2.0f / 1001.0f)

typedef __attribute__((ext_vector_type(2))) float v2f;
typedef __attribute__((ext_vector_type(8))) float v8f;

// One K-step body: loads 2 batch rows (K=ko, ko+1) for this lane's v column,
// builds one-hot A / attr B slices, runs one V_WMMA_F32_16X16X4_F32.
// Incremental pointers: sp walks sparse rows, wp walks W rows (stride 4 rows/iter).
__device__ __forceinline__ void wmma_step(const float*& sp, const float*& wp,
                                          const int* slab, int& bidx,
                                          int m, v8f& acc, bool do_prefetch) {
    const int la = slab[bidx];
    const int lc = slab[bidx + 1];

    // attr[b, v] = |sparse[b,v] * W_eff[b,v,labels[b]]| ; W gather stride = C floats.
    const float a0 = __builtin_nontemporal_load(sp);                         // row ko
    const float a1 = __builtin_nontemporal_load(sp + VV);                    // row ko+1 (131072B imm)
    const float w0 = __builtin_nontemporal_load(wp + la);                    // row ko, col la
    const float w1 = __builtin_nontemporal_load(wp + (size_t)VV * CC + lc);  // row ko+1 (2MB imm)

    if (do_prefetch) {
        // Next iteration's W rows (wp row base is 64B-aligned; covers any label col).
        __builtin_prefetch(wp + 4 * (size_t)VV * CC, 0, 1);
        __builtin_prefetch(wp + 5 * (size_t)VV * CC, 0, 1);
    }

    const v2f Bm = {fabsf(a0 * w0), fabsf(a1 * w1)};
    const v2f Am = {(la == m) ? 1.0f : 0.0f, (lc == m) ? 1.0f : 0.0f};

    acc = __builtin_amdgcn_wmma_f32_16x16x4_f32(
        /*neg_a=*/false, Am, /*neg_b=*/false, Bm,
        /*c_mod=*/(short)0, acc, /*reuse_a=*/false, /*reuse_b=*/false);

    sp += 4 * (size_t)VV;        // advance 4 batch rows
    wp += 4 * (size_t)VV * CC;
    bidx += 4;
}

// Kernel 1: partial segment-sums via f32 WMMA.
//   ws[split][c][v] = sum over b in split of onehot(labels[b])[c] * |sparse[b,v]*W[b,v,labels[b]]|
// One wave owns a 16(classes) x 16(v) f32 accumulator tile (8 VGPRs).
__global__ __launch_bounds__(256) void attr_centroid_partial_kernel(
    const float* __restrict__ sparse,   // [B, V]
    const float* __restrict__ W,        // [B, V, C]
    const int*   __restrict__ labels,   // [B]
    float*       __restrict__ ws,       // [splits, C, V]
    int rows_per_split)
{
    __shared__ int slab[BB];  // stage labels once per block (2 KB)
    const int tid = threadIdx.x;
    for (int b = tid; b < BB; b += 256) slab[b] = labels[b];
    __syncthreads();

    const int wave = tid >> 5;
    const int lane = tid & 31;
    const int n    = lane & 15;          // N column (B/C/D) and M row (A) for this lane
    const int hi   = lane >> 4;          // 0: K={0,1}, 1: K={2,3}
    const int ko   = hi << 1;

    const int v      = blockIdx.x * 128 + wave * 16 + n;   // this lane's v column
    const int b0base = blockIdx.y * rows_per_split;

    // Incremental base pointers for batch row (b0base + ko), column v.
    const size_t i0 = (size_t)(b0base + ko) * VV + v;
    const float* sp = sparse + i0;
    const float* wp = W + i0 * CC;
    int bidx = b0base + ko;

    v8f acc = {0.f, 0.f, 0.f, 0.f, 0.f, 0.f, 0.f, 0.f};

    // Main loop with prefetch; last K-chunk peeled (no prefetch, no in-loop branch).
    #pragma unroll 4
    for (int bb = 0; bb < rows_per_split - 4; bb += 4)
        wmma_step(sp, wp, slab, bidx, n, acc, /*do_prefetch=*/true);
    wmma_step(sp, wp, slab, bidx, n, acc, /*do_prefetch=*/false);

    // C/D layout: VGPR i -> M = i (lanes 0-15) / M = i+8 (lanes 16-31), N = lane&15
    float* wsBase = ws + (size_t)blockIdx.y * CC * VV;
    #pragma unroll
    for (int i = 0; i < 8; ++i) {
        const int m = i + (hi << 3);
        wsBase[(size_t)m * VV + v] = acc[i];
    }
}

// Kernel 2: reduce partials, compute mean, EMA / copy / keep selection.
__global__ __launch_bounds__(256) void attr_centroid_finalize_kernel(
    const float*         __restrict__ ws,          // [splits, C, V]
    const int*           __restrict__ labels,      // [B]
    const float*         __restrict__ centroids,   // [C, V]
    const unsigned char* __restrict__ initialized, // [C] (bool)
    float*               __restrict__ out,         // [C, V]
    int nsplits)
{
    __shared__ int cnt[CC];
    const int tid = threadIdx.x;
    if (tid < CC) cnt[tid] = 0;
    __syncthreads();
    for (int b = tid; b < BB; b += 256) atomicAdd(&cnt[labels[b]], 1);
    __syncthreads();

    const size_t idx = (size_t)blockIdx.x * 256 + tid;   // < C*V
    const int c = (int)(idx >> 15);                      // V = 2^15
    float s = 0.0f;
    for (int sp = 0; sp < nsplits; ++sp)
        s += ws[((size_t)sp * CC + c) * VV + (idx & (VV - 1))];

    const float count = (float)cnt[c];
    const float mean  = s / fmaxf(count, 1.0f);
    const float cen   = centroids[idx];
    const float lerp  = cen + ONE_MINUS_M * (mean - cen);
    const float upd   = initialized[c] ? lerp : mean;
    out[idx] = (count > 0.0f) ? upd : cen;
}

extern "C" void kernel_launch(void* const* d_in, const int* in_sizes, int n_in,
                              void* d_out, int out_size, void* d_ws, size_t ws_size,
                              hipStream_t stream) {
    const float*         sparse      = (const float*)d_in[0];          // [B,V]
    const float*         W           = (const float*)d_in[1];          // [B,V,C]
    const int*           labels      = (const int*)d_in[2];            // [B]
    const float*         centroids   = (const float*)d_in[3];          // [C,V]
    const unsigned char* initialized = (const unsigned char*)d_in[4];  // [C] bool
    float*               out         = (float*)d_out;                  // [C,V]
    float*               ws          = (float*)d_ws;                   // [splits,C,V]

    // Pick batch-split count that fits the workspace (each split = C*V floats = 2 MB).
    const size_t per_split = (size_t)CC * VV * sizeof(float);
    int splits = 8;
    while (splits > 1 && (size_t)splits * per_split > ws_size) splits >>= 1;
    const int rows_per_split = BB / splits;

    dim3 grid1(VV / 128, splits);       // 8 waves/block, 16 v-cols per wave
    attr_centroid_partial_kernel<<<grid1, 256, 0, stream>>>(
        sparse, W, labels, ws, rows_per_split);

    attr_centroid_finalize_kernel<<<(CC * VV) / 256, 256, 0, stream>>>(
        ws, labels, centroids, initialized, out, splits);
}